// BiChannelAttention_41291815584014
// MI455X (gfx1250) — compile-verified
//
#include <hip/hip_runtime.h>
#include <hip/hip_bf16.h>
#include <math.h>

// ---------------------------------------------------------------------------
// MI455X (gfx1250) optimized decode attention.
//
// Roofline: after exploiting the LOCAL_WINDOW=256 hard mask (scores before
// S-256 get -1e6 -> exp underflows to exactly 0 in fp32) and collapsing
//   scores = X * (Wk q),   out = (attn^T X) * Wv + bv + cnt,
// the problem is ~30 MFLOP against ~19 MB of traffic -> memory bound,
// ~1 us at 23.3 TB/s. The remaining dense GEMMs (q, kappa, out projections,
// each (16x128)x(128x128) per head) run on V_WMMA_F32_16X16X4_F32 (wave32).
// ---------------------------------------------------------------------------

typedef __attribute__((ext_vector_type(2))) float v2f;
typedef __attribute__((ext_vector_type(8))) float v8f;

#define BB 8
#define HH 16
#define HD 128
#define TT 2048
#define SS 2049
#define WIN 256
#define CUTOFF (SS - WIN)      // 1793
#define DD (HH * HD)           // 2048
#define NEGV (-1000000.0f)
#define LDS_STRIDE 132         // 128 + 4 pad: kills bank conflicts on frag loads

__device__ __forceinline__ float waveSum(float v) {
#pragma unroll
  for (int o = 16; o > 0; o >>= 1) v += __shfl_xor(v, o, 32);
  return v;
}
__device__ __forceinline__ float waveMax(float v) {
#pragma unroll
  for (int o = 16; o > 0; o >>= 1) v = fmaxf(v, __shfl_xor(v, o, 32));
  return v;
}

// ---------------------------------------------------------------------------
// Kernel 1: per head h.  q[b,:] = cnt[b,:]*Wq + bq  (WMMA, M=16 w/ 8 valid)
//           kappa[b,:]  = q[b,:] * Wk^T            (WMMA, B read transposed)
// WMMA f32 16x16x4 layouts (ISA 7.12.2):
//   A: lane l (l<16): M=l, K=k0+{0,1}; lane l>=16: M=l-16, K=k0+{2,3}
//   B: symmetric with N in place of M.  C/D: vgpr r -> M=r (lanes<16) / r+8.
// ---------------------------------------------------------------------------
__global__ void __launch_bounds__(256)
proj_qk_kernel(const float* __restrict__ content_t,
               const float* __restrict__ Wq, const float* __restrict__ bq,
               const float* __restrict__ Wk, float* __restrict__ kappa) {
  __shared__ __align__(16) float Alds[16 * LDS_STRIDE];
  __shared__ __align__(16) float Qlds[16 * LDS_STRIDE];
  const int tid = threadIdx.x, lane = tid & 31, wave = tid >> 5;
  const int half = lane >> 4, l16 = lane & 15;
  const int h = blockIdx.x;
  const float* Wq_h = Wq + h * HD * HD;
  const float* Wk_h = Wk + h * HD * HD;

  // stage cnt rows (8 valid, pad rows 8..15 with 0)
  for (int idx = tid; idx < 16 * HD; idx += 256) {
    int r = idx >> 7, c = idx & 127;
    Alds[r * LDS_STRIDE + c] = (r < 8) ? content_t[r * DD + h * HD + c] : 0.0f;
  }
  __syncthreads();

  const int n0 = wave * 16;                 // each wave owns one 16-col tile
  v8f accq = {};
  for (int k0 = 0; k0 < HD; k0 += 4) {
    const int ka = k0 + 2 * half;
    v2f a, bm;
    a.x = Alds[l16 * LDS_STRIDE + ka];
    a.y = Alds[l16 * LDS_STRIDE + ka + 1];
    bm.x = Wq_h[ka * HD + n0 + l16];        // B[k][n] = Wq[d=k][e=n]
    bm.y = Wq_h[(ka + 1) * HD + n0 + l16];
    accq = __builtin_amdgcn_wmma_f32_16x16x4_f32(false, a, false, bm,
                                                 (short)0, accq, false, false);
  }
  const float bqv = bq[h * HD + n0 + l16];
#pragma unroll
  for (int r = 0; r < 8; r++) accq[r] += bqv;
#pragma unroll
  for (int r = 0; r < 8; r++)
    Qlds[(r + 8 * half) * LDS_STRIDE + n0 + l16] = accq[r];
  __syncthreads();

  // kappa[b,d] = sum_e q[b,e] * Wk[d,e]   (B fragment: Wk transposed access)
  v8f acck = {};
  for (int k0 = 0; k0 < HD; k0 += 4) {
    const int ka = k0 + 2 * half;
    v2f a, bm;
    a.x = Qlds[l16 * LDS_STRIDE + ka];
    a.y = Qlds[l16 * LDS_STRIDE + ka + 1];
    bm.x = Wk_h[(n0 + l16) * HD + ka];      // B[k=e][n=d] = Wk[d][e]
    bm.y = Wk_h[(n0 + l16) * HD + ka + 1];
    acck = __builtin_amdgcn_wmma_f32_16x16x4_f32(false, a, false, bm,
                                                 (short)0, acck, false, false);
  }
  if (half == 0) {                          // rows 0..7 = valid batches
#pragma unroll
    for (int r = 0; r < 8; r++)
      kappa[(r * HH + h) * HD + n0 + l16] = acck[r];
  }
}

// ---------------------------------------------------------------------------
// Kernel 2: per (b,h).  scores over the 256-row window, softmax, y = attn^T X.
// Pass 1: one wave per 32 contiguous rows; lanes split the 128 cols as float4.
// Pass 2: threads map to columns (coalesced 128B/wave), rows split in half.
// ---------------------------------------------------------------------------
__global__ void __launch_bounds__(256)
attn_window_kernel(const float* __restrict__ content_t,
                   const int* __restrict__ time_mask,
                   const float* __restrict__ cache,
                   const float* __restrict__ kappa,
                   const float* __restrict__ pos_param,
                   float* __restrict__ y_ws) {
  __shared__ __align__(16) float kap[HD];
  __shared__ float attn_s[WIN];
  __shared__ float redmax[8], redsum[8];
  __shared__ float ypart[HD];

  const int tid = threadIdx.x, lane = tid & 31, wave = tid >> 5;
  const int b = blockIdx.x >> 4, h = blockIdx.x & 15;
  const float* cbase = cache + ((size_t)(b * TT + CUTOFF)) * DD + h * HD;
  const float* cntrow = content_t + (size_t)b * DD + h * HD;

  if (tid < HD) kap[tid] = kappa[(b * HH + h) * HD + tid];
  __syncthreads();

  auto rowPtr = [&](int sl) -> const float* {
    return (CUTOFF + sl < TT) ? (cbase + (size_t)sl * DD) : cntrow;
  };

  // ---- pass 1: scores (wave `wave` handles rows wave*32 .. wave*32+31) ----
  const float4 kv = ((const float4*)kap)[lane];
  float myScore = 0.0f;
  for (int i = 0; i < 32; i++) {
    const float* row = rowPtr(wave * 32 + i);
    if (i + 4 < 32)
      __builtin_prefetch(rowPtr(wave * 32 + i + 4) + 4 * lane, 0, 0);
    const float4 xv = ((const float4*)row)[lane];
    float p = xv.x * kv.x + xv.y * kv.y + xv.z * kv.z + xv.w * kv.w;
    p = waveSum(p);
    if (lane == i) myScore = p;
  }

  // per-thread row: sl = wave*32 + lane
  const int sl = wave * 32 + lane;
  const int n = (WIN - 1) - sl;             // distance to query
  float pos;
  if (n < 16) {
    pos = (float)n;
  } else {
    int large = 16 + (int)(logf((float)n * (1.0f / 16.0f)) *
                           (16.0f / logf(8.0f)));
    pos = (float)(large > 31 ? 31 : large);
  }
  const int tm = time_mask[b * SS + CUTOFF + sl];
  const float maskv = (tm == 1) ? 0.0f : NEGV;
  const float scale = 0.08838834764831845f;             // 1/sqrt(128)
  float sc = myScore * scale - pos_param[0] * pos + maskv;

  // ---- softmax over 256 ----
  float wmax = waveMax(sc);
  if (lane == 0) redmax[wave] = wmax;
  __syncthreads();
  float bmax = redmax[0];
#pragma unroll
  for (int j = 1; j < 8; j++) bmax = fmaxf(bmax, redmax[j]);
  const float e = __expf(sc - bmax);
  float wsum = waveSum(e);
  if (lane == 0) redsum[wave] = wsum;
  __syncthreads();
  float bsum = 0.0f;
#pragma unroll
  for (int j = 0; j < 8; j++) bsum += redsum[j];
  attn_s[sl] = e / bsum;
  __syncthreads();

  // ---- pass 2: y[c] = sum_s attn[s] * X[s][c]  (X re-read, hits L2) ----
  const int c = tid & 127;
  const int rbase = (tid >> 7) * 128;
  float acc = 0.0f;
  for (int r = 0; r < 128; r++) {
    const int sl2 = rbase + r;
    acc = fmaf(attn_s[sl2], rowPtr(sl2)[c], acc);
  }
  if (tid >= 128) ypart[c] = acc;
  __syncthreads();
  if (tid < 128) y_ws[(b * HH + h) * HD + tid] = acc + ypart[tid];
}

// ---------------------------------------------------------------------------
// Kernel 3: per head h.  out[b,:] = y[b,:]*Wv + bv + cnt[b,:]  (WMMA)
// ---------------------------------------------------------------------------
__global__ void __launch_bounds__(256)
proj_out_kernel(const float* __restrict__ y_ws,
                const float* __restrict__ Wv, const float* __restrict__ bv,
                const float* __restrict__ content_t, float* __restrict__ out) {
  __shared__ __align__(16) float Ylds[16 * LDS_STRIDE];
  const int tid = threadIdx.x, lane = tid & 31, wave = tid >> 5;
  const int half = lane >> 4, l16 = lane & 15;
  const int h = blockIdx.x;
  const float* Wv_h = Wv + h * HD * HD;

  for (int idx = tid; idx < 16 * HD; idx += 256) {
    int r = idx >> 7, c = idx & 127;
    Ylds[r * LDS_STRIDE + c] = (r < 8) ? y_ws[(r * HH + h) * HD + c] : 0.0f;
  }
  __syncthreads();

  const int n0 = wave * 16;
  v8f acc = {};
  for (int k0 = 0; k0 < HD; k0 += 4) {
    const int ka = k0 + 2 * half;
    v2f a, bm;
    a.x = Ylds[l16 * LDS_STRIDE + ka];
    a.y = Ylds[l16 * LDS_STRIDE + ka + 1];
    bm.x = Wv_h[ka * HD + n0 + l16];        // B[k=d][n=e] = Wv[d][e]
    bm.y = Wv_h[(ka + 1) * HD + n0 + l16];
    acc = __builtin_amdgcn_wmma_f32_16x16x4_f32(false, a, false, bm,
                                               (short)0, acc, false, false);
  }
  if (half == 0) {
    const float bvv = bv[h * HD + n0 + l16];
#pragma unroll
    for (int r = 0; r < 8; r++) {
      const int o = r * DD + h * HD + n0 + l16;
      out[o] = acc[r] + bvv + content_t[o];
    }
  }
}

// ---------------------------------------------------------------------------
extern "C" void kernel_launch(void* const* d_in, const int* in_sizes, int n_in,
                              void* d_out, int out_size, void* d_ws,
                              size_t ws_size, hipStream_t stream) {
  const float* content_t = (const float*)d_in[1];
  const int* time_mask = (const int*)d_in[2];
  const float* cache = (const float*)d_in[3];
  const float* Wq = (const float*)d_in[5];
  const float* bq = (const float*)d_in[6];
  const float* Wk = (const float*)d_in[7];
  // d_in[8] = bk: provably irrelevant (constant score shift, softmax-invariant)
  const float* Wv = (const float*)d_in[9];
  const float* bv = (const float*)d_in[10];
  const float* pos_param = (const float*)d_in[11];
  float* out = (float*)d_out;

  float* kappa = (float*)d_ws;            // B*H*HD floats = 64 KB
  float* y_ws = kappa + BB * HH * HD;     // B*H*HD floats = 64 KB

  proj_qk_kernel<<<HH, 256, 0, stream>>>(content_t, Wq, bq, Wk, kappa);
  attn_window_kernel<<<BB * HH, 256, 0, stream>>>(content_t, time_mask, cache,
                                                  kappa, pos_param, y_ws);
  proj_out_kernel<<<HH, 256, 0, stream>>>(y_ws, Wv, bv, content_t, out);
}